// InnerProduct_24240795418640
// MI455X (gfx1250) — compile-verified
//
#include <hip/hip_runtime.h>

// ---- problem constants ----
#define B_SZ   8192
#define N_F    64
#define M_E    32
#define KX     2048          // N_F * M_E
#define D_P    2016
#define OUT_F  1024
#define K_AUG  2112          // KX + N_F
#define AFF_K  2080          // N_F + D_P
#define BN_EPS 1e-5f

typedef __attribute__((ext_vector_type(2))) float v2f;
typedef __attribute__((ext_vector_type(8))) float v8f;

// ------------------------------------------------------------------
// Xsq[b,n] = sum_m X[b, n*32+m]^2     (idx = b*64+n, contiguous 32 floats)
// ------------------------------------------------------------------
__global__ void k_xsq(const float* __restrict__ X, float* __restrict__ Xsq) {
    int idx = blockIdx.x * blockDim.x + threadIdx.x;
    if (idx >= B_SZ * N_F) return;
    const float4* p = (const float4*)(X + (size_t)idx * M_E);
    float s = 0.f;
#pragma unroll
    for (int i = 0; i < M_E / 4; ++i) {
        float4 v = p[i];
        s += v.x * v.x + v.y * v.y + v.z * v.z + v.w * v.w;
    }
    Xsq[idx] = s;
}

// Pair-interleaved weight layout: element (k,o) lives at
//   WaugP[ ((k>>1)*OUT_F + o)*2 + (k&1) ]
// so a {k, k+1} fragment pair (k even) is one contiguous float2.
__device__ __forceinline__ size_t waug_idx(int k, int o) {
    return ((size_t)(k >> 1) * OUT_F + o) * 2 + (k & 1);
}

// ------------------------------------------------------------------
// Waug[k,o] (k<2048)  = sum_d Wz[d,k] * Wa[o,d]
// ------------------------------------------------------------------
__global__ void k_build_w1(const float* __restrict__ Wz,
                           const float* __restrict__ Wa,
                           float* __restrict__ WaugP) {
    int idx = blockIdx.x * blockDim.x + threadIdx.x;   // over KX*OUT
    if (idx >= KX * OUT_F) return;
    int k = idx / OUT_F;
    int o = idx % OUT_F;
    float s = 0.f;
#pragma unroll 8
    for (int d = 0; d < N_F; ++d)
        s += Wz[(size_t)d * KX + k] * Wa[(size_t)o * AFF_K + d];
    WaugP[waug_idx(k, o)] = s;
}

// ------------------------------------------------------------------
// Waug[2048+n,o] = sum_j theta[j,n]^2 * Wa[o, 64+j]
// ------------------------------------------------------------------
__global__ void k_build_w2(const float* __restrict__ theta,
                           const float* __restrict__ Wa,
                           float* __restrict__ WaugP) {
    int idx = blockIdx.x * blockDim.x + threadIdx.x;   // over N_F*OUT
    if (idx >= N_F * OUT_F) return;
    int n = idx / OUT_F;
    int o = idx % OUT_F;
    float s = 0.f;
    for (int j = 0; j < D_P; ++j) {
        float t = theta[(size_t)j * N_F + n];
        s += t * t * Wa[(size_t)o * AFF_K + N_F + j];
    }
    WaugP[waug_idx(KX + n, o)] = s;
}

// ------------------------------------------------------------------
// Main fused GEMM: Out[b,o] = sum_k Aaug[b,k]*Waug[k,o] + bias[o]
// Aaug row = [ X row (2048) | Xsq row (64) ]
// Block: 256 thr = 8 waves; tile 32 rows x 512 cols.
// Each wave: 2 row-tiles x 4 col-tiles = 8 WMMA f32 16x16x4 per k-step,
// B fragment (float2, pair-interleaved) shared by both row-tiles.
// A staged through double-buffered LDS, one barrier per K-chunk.
// ------------------------------------------------------------------
#define KC     64
#define NCHUNK (K_AUG / KC)   // 33

__global__ __launch_bounds__(256) void k_gemm(const float* __restrict__ X,
                                              const float* __restrict__ Xsq,
                                              const float* __restrict__ WaugP,
                                              const float* __restrict__ bias,
                                              float* __restrict__ Out) {
    __shared__ float At[2][32][KC + 4];   // +4 pad: conflict-free, 16B-aligned rows

    const int tid  = threadIdx.x;
    const int wave = tid >> 5;
    const int lane = tid & 31;
    const int row0 = blockIdx.x * 32;
    const int col0 = blockIdx.y * 512 + wave * 64;
    const int r    = lane & 15;       // tile row (A) / tile col (B,D)
    const int kh   = lane >> 4;       // K-half select per ISA layout

    v8f acc[2][4];
    const v8f vzero = {0.f, 0.f, 0.f, 0.f, 0.f, 0.f, 0.f, 0.f};
#pragma unroll
    for (int rt = 0; rt < 2; ++rt)
#pragma unroll
        for (int t = 0; t < 4; ++t) acc[rt][t] = vzero;

    const int lr = tid >> 4;          // 0..15  staging row
    const int lc = (tid & 15) * 4;    // 0..60  staging col (float4)

    auto src_ptr = [&](int c, int rowoff) -> const float* {
        const int k0 = c * KC;
        return (k0 < KX)
            ? (X   + (size_t)(row0 + rowoff) * KX  + (k0      + lc))
            : (Xsq + (size_t)(row0 + rowoff) * N_F + (k0 - KX + lc));
    };

    // prologue: stage chunk 0
    *(float4*)&At[0][lr     ][lc] = *(const float4*)src_ptr(0, lr);
    *(float4*)&At[0][lr + 16][lc] = *(const float4*)src_ptr(0, lr + 16);
    __syncthreads();

    for (int c = 0; c < NCHUNK; ++c) {
        const int cur = c & 1, nxt = cur ^ 1;
        const bool have = (c + 1) < NCHUNK;
        float4 n0, n1;
        if (have) {                       // issue next chunk's global loads early
            n0 = *(const float4*)src_ptr(c + 1, lr);
            n1 = *(const float4*)src_ptr(c + 1, lr + 16);
        }

        const int k0 = c * KC;
#pragma unroll 4
        for (int kk = 0; kk < KC; kk += 4) {
            // A fragments: lanes 0-15 hold {K0,K1}, lanes 16-31 hold {K2,K3}
            v2f a0, a1;
            a0.x = At[cur][r     ][kk + 2 * kh];
            a0.y = At[cur][r     ][kk + 2 * kh + 1];
            a1.x = At[cur][r + 16][kk + 2 * kh];
            a1.y = At[cur][r + 16][kk + 2 * kh + 1];
            const int kg = k0 + kk + 2 * kh;           // always even
            const v2f* bp = (const v2f*)WaugP + (size_t)(kg >> 1) * OUT_F + col0 + r;
#pragma unroll
            for (int t = 0; t < 4; ++t) {
                const v2f b = bp[t * 16];              // one b64 load per fragment
                acc[0][t] = __builtin_amdgcn_wmma_f32_16x16x4_f32(
                    false, a0, false, b, (short)0, acc[0][t], false, false);
                acc[1][t] = __builtin_amdgcn_wmma_f32_16x16x4_f32(
                    false, a1, false, b, (short)0, acc[1][t], false, false);
            }
        }

        if (have) {                       // fill the other buffer
            *(float4*)&At[nxt][lr     ][lc] = n0;
            *(float4*)&At[nxt][lr + 16][lc] = n1;
        }
        __syncthreads();
    }

    // store D (+bias): VGPR v, lanes 0-15 -> M=v, lanes 16-31 -> M=v+8
#pragma unroll
    for (int rt = 0; rt < 2; ++rt) {
#pragma unroll
        for (int t = 0; t < 4; ++t) {
            const int col = col0 + t * 16 + r;
            const float bv = bias[col];
#pragma unroll
            for (int v = 0; v < 8; ++v) {
                const int row = row0 + rt * 16 + v + 8 * kh;
                Out[(size_t)row * OUT_F + col] = acc[rt][t][v] + bv;
            }
        }
    }
}

// ------------------------------------------------------------------
// BatchNorm: deterministic two-level reduction (no float atomics)
// ------------------------------------------------------------------
__global__ void k_bn_partial(const float* __restrict__ Out,
                             float* __restrict__ pSum, float* __restrict__ pSq) {
    const int chunk = blockIdx.x;          // 32 row-chunks of 256 rows
    const int tid   = threadIdx.x;         // 256 threads
    const int r0    = chunk * 256;
    float s[4] = {0, 0, 0, 0}, q[4] = {0, 0, 0, 0};
    for (int rr = 0; rr < 256; ++rr) {
        const float* row = Out + (size_t)(r0 + rr) * OUT_F;
#pragma unroll
        for (int f = 0; f < 4; ++f) {
            float v = row[f * 256 + tid];
            s[f] += v;
            q[f] += v * v;
        }
    }
#pragma unroll
    for (int f = 0; f < 4; ++f) {
        pSum[chunk * OUT_F + f * 256 + tid] = s[f];
        pSq [chunk * OUT_F + f * 256 + tid] = q[f];
    }
}

__global__ void k_bn_stats(const float* __restrict__ pSum,
                           const float* __restrict__ pSq,
                           float* __restrict__ mean, float* __restrict__ inv) {
    int o = blockIdx.x * blockDim.x + threadIdx.x;
    if (o >= OUT_F) return;
    float s = 0.f, q = 0.f;
    for (int c = 0; c < 32; ++c) {
        s += pSum[c * OUT_F + o];
        q += pSq [c * OUT_F + o];
    }
    const float m   = s / (float)B_SZ;
    const float var = q / (float)B_SZ - m * m;   // biased variance
    mean[o] = m;
    inv[o]  = rsqrtf(var + BN_EPS);
}

__global__ void k_bn_apply(float* __restrict__ Out,
                           const float* __restrict__ mean, const float* __restrict__ inv,
                           const float* __restrict__ gamma, const float* __restrict__ beta) {
    int i = blockIdx.x * blockDim.x + threadIdx.x;
    if (i >= B_SZ * OUT_F) return;
    int o = i & (OUT_F - 1);
    Out[i] = (Out[i] - mean[o]) * inv[o] * gamma[o] + beta[o];
}

// ------------------------------------------------------------------
extern "C" void kernel_launch(void* const* d_in, const int* in_sizes, int n_in,
                              void* d_out, int out_size, void* d_ws, size_t ws_size,
                              hipStream_t stream) {
    const float* X     = (const float*)d_in[0];   // [B, 2048]
    const float* Wz    = (const float*)d_in[1];   // [64, 64, 32]
    const float* theta = (const float*)d_in[2];   // [2016, 64]
    const float* Wa    = (const float*)d_in[3];   // [1024, 2080]
    const float* bAff  = (const float*)d_in[4];   // [1024]
    const float* gamma = (const float*)d_in[5];   // [1024]
    const float* beta  = (const float*)d_in[6];   // [1024]
    float* Out = (float*)d_out;                   // [8192, 1024]

    float* ws    = (float*)d_ws;
    float* Xsq   = ws;                                 // 8192*64      = 524288
    float* WaugP = Xsq   + (size_t)B_SZ * N_F;         // 2112*1024    = 2162688
    float* pSum  = WaugP + (size_t)K_AUG * OUT_F;      // 32*1024
    float* pSq   = pSum  + 32 * OUT_F;                 // 32*1024
    float* mean  = pSq   + 32 * OUT_F;                 // 1024
    float* inv   = mean  + OUT_F;                      // 1024

    k_xsq<<<(B_SZ * N_F) / 256, 256, 0, stream>>>(X, Xsq);
    k_build_w1<<<(KX * OUT_F) / 256, 256, 0, stream>>>(Wz, Wa, WaugP);
    k_build_w2<<<(N_F * OUT_F) / 256, 256, 0, stream>>>(theta, Wa, WaugP);

    dim3 gGemm(B_SZ / 32, OUT_F / 512);
    k_gemm<<<gGemm, 256, 0, stream>>>(X, Xsq, WaugP, bAff, Out);

    k_bn_partial<<<32, 256, 0, stream>>>(Out, pSum, pSq);
    k_bn_stats<<<(OUT_F + 255) / 256, 256, 0, stream>>>(pSum, pSq, mean, inv);
    k_bn_apply<<<(B_SZ * OUT_F) / 256, 256, 0, stream>>>(Out, mean, inv, gamma, beta);
}